// HypAgg_21131239097070
// MI455X (gfx1250) — compile-verified
//
#include <hip/hip_runtime.h>
#include <hip/hip_bf16.h>

// HypAgg (Poincare ball aggregation), B=1, N=1024, D=128, c=1.
// Algebraic collapse: pairwise logmap is a linear combo of x_i and x_j whose
// coefficients depend only on Gram(i,j), |x_i|^2, |x_j|^2. So:
//   support_t = S1 .* X  +  P @ X,   with G = X @ X^T computed by WMMA f32.
// Two fp32 WMMA GEMMs (1024x1024x128 each) + O(N^2) VALU transcendentals.

#define N_PTS 1024
#define D_DIM 128
#define MINN  1e-15f
#define EPSC  1e-7f

typedef __attribute__((ext_vector_type(2))) float v2f;
typedef __attribute__((ext_vector_type(8))) float v8f;

// ---------------------------------------------------------------- kernel 1 --
// per point: x2[i]=|x_i|^2 ; logmap0 factor f=atanh(|x|)/|x| ;
// left[i]=f*(x_i . w[:D]) ; right[i]=f*(x_i . w[D:])
__global__ void k_point_stats(const float* __restrict__ x,
                              const float* __restrict__ att_w,
                              float* __restrict__ x2v,
                              float* __restrict__ leftv,
                              float* __restrict__ rightv) {
  const int lane = threadIdx.x & 31;
  const int wid  = threadIdx.x >> 5;
  const int i    = blockIdx.x * 4 + wid;          // 4 waves/block
  float4 v  = ((const float4*)(x + i * D_DIM))[lane];
  float4 wl = ((const float4*)(att_w))[lane];
  float4 wr = ((const float4*)(att_w + D_DIM))[lane];
  float sxx = v.x*v.x + v.y*v.y + v.z*v.z + v.w*v.w;
  float swl = v.x*wl.x + v.y*wl.y + v.z*wl.z + v.w*wl.w;
  float swr = v.x*wr.x + v.y*wr.y + v.z*wr.z + v.w*wr.w;
#pragma unroll
  for (int off = 16; off >= 1; off >>= 1) {
    sxx += __shfl_xor(sxx, off, 32);
    swl += __shfl_xor(swl, off, 32);
    swr += __shfl_xor(swr, off, 32);
  }
  if (lane == 0) {
    x2v[i] = sxx;
    float n = sqrtf(fmaxf(sxx, MINN));
    float u = fminf(n, 1.0f - EPSC);
    float f = atanhf(u) / n;                      // logmap0 scale
    leftv[i]  = f * swl;
    rightv[i] = f * swr;
  }
}

// ---------------------------------------------------------------- kernel 2 --
// G = X @ X^T  via V_WMMA_F32_16X16X4_F32. One wave per 16x16 tile.
// A-frag (ISA 7.12.2): lanes 0-15 -> M=lane, K=k..k+1 ; lanes 16-31 -> K=k+2..k+3.
// B-frag assumed mirrored (N=lane&15, K split 0,1 | 2,3 by half-wave); for the
// symmetric Gram both fragments use the identical load pattern.
__global__ void k_gram(const float* __restrict__ x, float* __restrict__ G) {
  const int lane = threadIdx.x & 31;
  const int wid  = threadIdx.x >> 5;
  const int i0 = blockIdx.y * 16;
  const int j0 = (blockIdx.x * 4 + wid) * 16;
  const int half = lane >> 4;
  const int l    = lane & 15;
  const float* arow = x + (i0 + l) * D_DIM + half * 2;
  const float* brow = x + (j0 + l) * D_DIM + half * 2;
  v8f c = {0.f,0.f,0.f,0.f,0.f,0.f,0.f,0.f};
#pragma unroll 4
  for (int k = 0; k < D_DIM; k += 4) {
    v2f a = *(const v2f*)(arow + k);
    v2f b = *(const v2f*)(brow + k);
    c = __builtin_amdgcn_wmma_f32_16x16x4_f32(false, a, false, b,
                                              (short)0, c, false, false);
  }
  float* gout = G + (size_t)(i0 + half * 8) * N_PTS + j0 + l;
#pragma unroll
  for (int v = 0; v < 8; v++) gout[(size_t)v * N_PTS] = c[v];
}

// ---------------------------------------------------------------- kernel 3 --
// Per pair (i,j): den, alpha, beta, sn, w, att ->  P[i,j] = att*w*beta,
// S1[i] = sum_j att*w*alpha.  One block per row i.
__global__ void k_pairwise(const float* __restrict__ G,
                           const float* __restrict__ adj,
                           const float* __restrict__ x2v,
                           const float* __restrict__ leftv,
                           const float* __restrict__ rightv,
                           const float* __restrict__ att_b,
                           float* __restrict__ P,
                           float* __restrict__ S1) {
  const int i = blockIdx.x;
  const float xx2 = x2v[i];
  const float li  = leftv[i];
  const float bb  = att_b[0];
  const float one_m_x2 = fmaxf(1.0f - xx2, MINN);   // 2/lambda_i
  float s1 = 0.0f;
  for (int j = threadIdx.x; j < N_PTS; j += 256) {
    float d   = G[(size_t)i * N_PTS + j];
    float y2  = x2v[j];
    float den = fmaxf(1.0f - 2.0f * d + xx2 * y2, MINN);
    float alpha = -(1.0f - 2.0f * d + y2) / den;
    float beta  = (1.0f - xx2) / den;
    float sn2 = alpha * alpha * xx2 + 2.0f * alpha * beta * d + beta * beta * y2;
    float sn  = sqrtf(fmaxf(sn2, MINN));
    float w   = one_m_x2 * atanhf(fminf(sn, 1.0f - EPSC)) / sn;
    float z   = li + rightv[j] + bb;
    float att = adj[(size_t)i * N_PTS + j] / (1.0f + expf(-z));
    float aw  = att * w;
    s1 += aw * alpha;
    P[(size_t)i * N_PTS + j] = aw * beta;
  }
#pragma unroll
  for (int off = 16; off >= 1; off >>= 1) s1 += __shfl_xor(s1, off, 32);
  __shared__ float ws[8];
  if ((threadIdx.x & 31) == 0) ws[threadIdx.x >> 5] = s1;
  __syncthreads();
  if (threadIdx.x == 0) {
    float t = 0.f;
#pragma unroll
    for (int k = 0; k < 8; k++) t += ws[k];
    S1[i] = t;
  }
}

// ---------------------------------------------------------------- kernel 4 --
// support_t = P @ X + S1.*X  (WMMA f32), then fused expmap/mobius_add.
// Block = 8 waves; each wave owns one 16x16 column tile of a 16x128 row block.
__global__ void k_support_expmap(const float* __restrict__ P,
                                 const float* __restrict__ x,
                                 const float* __restrict__ x2v,
                                 const float* __restrict__ S1,
                                 float* __restrict__ out) {
  __shared__ float u_tile[16][D_DIM];
  __shared__ float part[16][8][2];
  __shared__ float coefA[16], coefB[16];

  const int tid  = threadIdx.x;
  const int lane = tid & 31;
  const int wid  = tid >> 5;              // 0..7 -> column tile
  const int i0   = blockIdx.x * 16;
  const int c0   = wid * 16;              // column base in D
  const int half = lane >> 4;
  const int l    = lane & 15;

  const float* arow  = P + (size_t)(i0 + l) * N_PTS + half * 2;
  const float* bbase = x + c0 + l;        // B[K,N] = x[K*128 + c0 + N]
  v8f c = {0.f,0.f,0.f,0.f,0.f,0.f,0.f,0.f};
  for (int k = 0; k < N_PTS; k += 4) {
    v2f a = *(const v2f*)(arow + k);
    v2f b;
    b.x = bbase[(size_t)(k + 2 * half)     * D_DIM];
    b.y = bbase[(size_t)(k + 2 * half + 1) * D_DIM];
    c = __builtin_amdgcn_wmma_f32_16x16x4_f32(false, a, false, b,
                                              (short)0, c, false, false);
  }
#pragma unroll
  for (int v = 0; v < 8; v++) {
    int r = v + half * 8;
    float val = c[v] + S1[i0 + r] * x[(size_t)(i0 + r) * D_DIM + c0 + l];
    u_tile[r][c0 + l] = val;
  }
  __syncthreads();

  // row reductions: |u|^2 and x.u  (16 rows x 8 segments of 16)
  if (tid < 128) {
    int r = tid >> 3, seg = tid & 7;
    const float* ur = &u_tile[r][seg * 16];
    const float* xr = x + (size_t)(i0 + r) * D_DIM + seg * 16;
    float suu = 0.f, sxu = 0.f;
#pragma unroll
    for (int d = 0; d < 16; d++) { float uv = ur[d]; suu += uv * uv; sxu += xr[d] * uv; }
    part[r][seg][0] = suu; part[r][seg][1] = sxu;
  }
  __syncthreads();

  if (tid < 16) {
    int r = tid;
    float uu = 0.f, xu = 0.f;
#pragma unroll
    for (int s = 0; s < 8; s++) { uu += part[r][s][0]; xu += part[r][s][1]; }
    float x2  = x2v[i0 + r];
    float un  = sqrtf(fmaxf(uu, MINN));
    float lam = 2.0f / fmaxf(1.0f - x2, MINN);
    float t   = tanhf(lam * un * 0.5f);
    float tv  = t / un;                    // second = tv * u
    float y2  = tv * tv * uu;
    float xy  = tv * xu;
    float den = fmaxf(1.0f + 2.0f * xy + x2 * y2, MINN);
    coefA[r] = (1.0f + 2.0f * xy + y2) / den;
    coefB[r] = (1.0f - x2) * tv / den;
  }
  __syncthreads();

  for (int e = tid; e < 16 * D_DIM; e += 256) {
    int r = e >> 7, d = e & (D_DIM - 1);
    out[(size_t)(i0 + r) * D_DIM + d] =
        coefA[r] * x[(size_t)(i0 + r) * D_DIM + d] + coefB[r] * u_tile[r][d];
  }
}

// ------------------------------------------------------------------ launch --
extern "C" void kernel_launch(void* const* d_in, const int* in_sizes, int n_in,
                              void* d_out, int out_size, void* d_ws, size_t ws_size,
                              hipStream_t stream) {
  const float* x     = (const float*)d_in[0];   // [1,1024,128]
  const float* adj   = (const float*)d_in[1];   // [1,1024,1024]
  const float* att_w = (const float*)d_in[2];   // [256]
  const float* att_b = (const float*)d_in[3];   // scalar
  float* out = (float*)d_out;                   // [1,1024,128]

  char* ws = (char*)d_ws;
  float* G     = (float*)(ws);                               // 4 MB
  float* Pm    = (float*)(ws + (size_t)N_PTS * N_PTS * 4);   // 4 MB
  float* x2v   = (float*)(ws + (size_t)2 * N_PTS * N_PTS * 4);
  float* leftv = x2v + N_PTS;
  float* rightv= leftv + N_PTS;
  float* S1    = rightv + N_PTS;

  k_point_stats<<<N_PTS / 4, 128, 0, stream>>>(x, att_w, x2v, leftv, rightv);
  k_gram<<<dim3(N_PTS / 64, N_PTS / 16), 128, 0, stream>>>(x, G);
  k_pairwise<<<N_PTS, 256, 0, stream>>>(G, adj, x2v, leftv, rightv, att_b, Pm, S1);
  k_support_expmap<<<N_PTS / 16, 256, 0, stream>>>(Pm, x, x2v, S1, out);
}